// BiasAndSum_32195074851337
// MI455X (gfx1250) — compile-verified
//
#include <hip/hip_runtime.h>

// Reference: xs (32768, 1, 2048) f32 -> (ys = xs+1, carry = colsum(ys))
// HBM-bound: ~512MB traffic -> ~22us floor at 23.3 TB/s.
// Phase 1: streaming bias + per-block partial column sums (b128 coalesced).
// Phase 2: carry = ones(1xP) x partials via V_WMMA_F32_16X16X4_F32 (exact f32).

#define D_COLS 2048

typedef __attribute__((ext_vector_type(2))) float v2f;
typedef __attribute__((ext_vector_type(8))) float v8f;

__global__ void __launch_bounds__(256)
bias_partial_kernel(const float* __restrict__ xs,
                    float* __restrict__ ys,
                    float* __restrict__ partial,
                    int T, int rows_per_block) {
    // D_COLS = 2048 floats = 512 float4 groups; 256 threads -> 2 groups/thread.
    const int g0 = threadIdx.x;
    const int g1 = threadIdx.x + 256;
    const int r0 = blockIdx.x * rows_per_block;
    int r1 = r0 + rows_per_block;
    if (r1 > T) r1 = T;

    float4 acc0 = make_float4(0.f, 0.f, 0.f, 0.f);
    float4 acc1 = make_float4(0.f, 0.f, 0.f, 0.f);

    for (int r = r0; r < r1; ++r) {
        const float4* xrow = reinterpret_cast<const float4*>(xs + (size_t)r * D_COLS);
        float4*       yrow = reinterpret_cast<float4*>(ys + (size_t)r * D_COLS);
        float4 a = xrow[g0];
        float4 b = xrow[g1];
        a.x += 1.f; a.y += 1.f; a.z += 1.f; a.w += 1.f;
        b.x += 1.f; b.y += 1.f; b.z += 1.f; b.w += 1.f;
        yrow[g0] = a;
        yrow[g1] = b;
        acc0.x += a.x; acc0.y += a.y; acc0.z += a.z; acc0.w += a.w;
        acc1.x += b.x; acc1.y += b.y; acc1.z += b.z; acc1.w += b.w;
    }

    float4* prow = reinterpret_cast<float4*>(partial + (size_t)blockIdx.x * D_COLS);
    prow[g0] = acc0;
    prow[g1] = acc1;
}

// Fallback streaming bias (no workspace needed).
__global__ void __launch_bounds__(256)
bias_only_kernel(const float4* __restrict__ x, float4* __restrict__ y, long n4) {
    long i = (long)blockIdx.x * blockDim.x + threadIdx.x;
    const long stride = (long)gridDim.x * blockDim.x;
    for (; i < n4; i += stride) {
        float4 v = x[i];
        v.x += 1.f; v.y += 1.f; v.z += 1.f; v.w += 1.f;
        y[i] = v;
    }
}

// carry[n] = sum over nrows of src[r][n], computed as ones(16x4) x B(4x16)
// accumulated with v_wmma_f32_16x16x4_f32. One wave (32 lanes) per 16-col tile.
// nrows must be a multiple of 4 (guaranteed by the launcher) so EXEC stays
// all-ones through the WMMA loop, as the ISA requires.
__global__ void __launch_bounds__(32)
wmma_colsum_kernel(const float* __restrict__ src,
                   float* __restrict__ carry,
                   int nrows) {
    const int lane = threadIdx.x;          // wave32
    const int tile = blockIdx.x;           // 16-column tile index
    const int col  = tile * 16 + (lane & 15);
    const int half = lane >> 4;            // 0: K rows {k,k+1}, 1: {k+2,k+3}

    v2f a;                                 // A-matrix: all ones (16x4)
    a[0] = 1.0f;
    a[1] = 1.0f;
    v8f c = {};                            // 16x16 f32 accumulator

    for (int k = 0; k < nrows; k += 4) {
        const int kr = k + 2 * half;
        v2f b;                             // B-matrix 4x16: VGPR0 holds K=kr,
        b[0] = src[(size_t)kr * D_COLS + col];        // VGPR1 holds K=kr+1
        b[1] = src[(size_t)(kr + 1) * D_COLS + col];
        // (neg_a, A, neg_b, B, c_mod, C, reuse_a, reuse_b)
        c = __builtin_amdgcn_wmma_f32_16x16x4_f32(
                false, a, false, b, (short)0, c, false, false);
    }

    // D layout: VGPR0, lanes 0-15 hold row M=0, N=lane -> exact column sums.
    if (lane < 16) {
        carry[col] = c[0];
    }
}

extern "C" void kernel_launch(void* const* d_in, const int* in_sizes, int n_in,
                              void* d_out, int out_size, void* d_ws, size_t ws_size,
                              hipStream_t stream) {
    const float* xs = (const float*)d_in[0];
    const int D = D_COLS;
    const int T = in_sizes[0] / D;         // 32768 for the reference shapes

    float* ys    = (float*)d_out;
    float* carry = ys + (size_t)T * D;

    // Number of partial rows we can hold in workspace (multiple of 4, <= 512).
    int P = (int)(ws_size / ((size_t)D * sizeof(float)));
    if (P > 512) P = 512;
    P &= ~3;

    if (P >= 4) {
        float* partial = (float*)d_ws;
        const int rpb = (T + P - 1) / P;
        bias_partial_kernel<<<dim3(P), dim3(256), 0, stream>>>(xs, ys, partial, T, rpb);
        wmma_colsum_kernel<<<dim3(D / 16), dim3(32), 0, stream>>>(partial, carry, P);
    } else {
        // Workspace too small: stream the bias, then WMMA-reduce ys directly
        // (T is a multiple of 4).
        const long n4 = (long)T * D / 4;
        bias_only_kernel<<<dim3(2048), dim3(256), 0, stream>>>(
            (const float4*)xs, (float4*)ys, n4);
        wmma_colsum_kernel<<<dim3(D / 16), dim3(32), 0, stream>>>(ys, carry, T);
    }
}